// CopyDecoder_29523605192796
// MI455X (gfx1250) — compile-verified
//
#include <hip/hip_runtime.h>
#include <hip/hip_bf16.h>

// ---------------- problem constants ----------------
constexpr int B    = 32;
constexpr int S    = 2048;
constexpr int H    = 256;
constexpr int E    = 256;
constexpr int V    = 50000;
constexpr int OOV  = 12;
constexpr int VEXT = V + OOV;      // 50012
constexpr int TWOH = 2 * H;        // 512
constexpr int XDIM = E + TWOH;     // 768

// ---------------- CDNA5 WMMA helpers ----------------
typedef __attribute__((ext_vector_type(2))) float v2f;
typedef __attribute__((ext_vector_type(8))) float v8f;
typedef __attribute__((ext_vector_type(4))) int   v4i;

// D(16x16,f32) = A(16x4,f32) x B(4x16,f32) + C  — wave32, 8-arg form
__device__ __forceinline__ v8f wmma4(v2f a, v2f b, v8f c) {
    return __builtin_amdgcn_wmma_f32_16x16x4_f32(
        /*neg_a=*/false, a, /*neg_b=*/false, b,
        /*c_mod=*/(short)0, c, /*reuse_a=*/false, /*reuse_b=*/false);
}

// ---------------- CDNA5 async global->LDS staging ----------------
#if defined(__AMDGCN__) && \
    __has_builtin(__builtin_amdgcn_global_load_async_to_lds_b128) && \
    __has_builtin(__builtin_amdgcn_s_wait_asynccnt)
#define USE_ASYNC_LDS 1
#else
#define USE_ASYNC_LDS 0
#endif

#define GLOBAL_AS __attribute__((address_space(1)))
#define LDS_AS    __attribute__((address_space(3)))

// Cooperative copy of `nfloat` floats (multiple of 1024) from global to LDS,
// 256 threads, 16B per thread per step. Includes the async-counter wait but
// NOT the block barrier (caller must __syncthreads()).
__device__ __forceinline__ void stage_to_lds(const float* __restrict__ gsrc,
                                             float* __restrict__ ldst, int nfloat) {
#if USE_ASYNC_LDS
    for (int i = threadIdx.x * 4; i < nfloat; i += 256 * 4) {
        __builtin_amdgcn_global_load_async_to_lds_b128(
            (GLOBAL_AS v4i*)(gsrc + i), (LDS_AS v4i*)(ldst + i), 0, 0);
    }
    __builtin_amdgcn_s_wait_asynccnt(0);
#else
    for (int i = threadIdx.x; i < nfloat; i += 256) ldst[i] = gsrc[i];
#endif
}

// ---------------- K0: zero scratch words ----------------
__global__ void zero_kernel(unsigned* __restrict__ p, long n) {
    long i = (long)blockIdx.x * blockDim.x + threadIdx.x;
    if (i < n) p[i] = 0u;
}

// ---------------- K1: GRU cell -> state ----------------
__global__ void __launch_bounds__(256)
gru_kernel(const int* __restrict__ input_idx, const float* __restrict__ embed,
           const float* __restrict__ weighted, const float* __restrict__ prev_state,
           const float* __restrict__ w_ih, const float* __restrict__ w_hh,
           const float* __restrict__ b_ih, const float* __restrict__ b_hh,
           float* __restrict__ state_out) {
    __shared__ float x[XDIM];
    __shared__ float h[H];
    const int b = blockIdx.x;
    const int t = threadIdx.x;                 // 0..255
    const int e = input_idx[b];
    x[t]         = embed[(long)e * E + t];
    x[E + t]     = weighted[b * TWOH + t];
    x[E + H + t] = weighted[b * TWOH + H + t];
    h[t]         = prev_state[b * H + t];
    __syncthreads();

    float gxr = b_ih[t], gxz = b_ih[H + t], gxn = b_ih[2 * H + t];
    const float* wr = w_ih + (long)t * XDIM;
    const float* wz = w_ih + (long)(H + t) * XDIM;
    const float* wn = w_ih + (long)(2 * H + t) * XDIM;
    for (int k = 0; k < XDIM; ++k) {
        float xv = x[k];
        gxr += wr[k] * xv; gxz += wz[k] * xv; gxn += wn[k] * xv;
    }
    float ghr = b_hh[t], ghz = b_hh[H + t], ghn = b_hh[2 * H + t];
    const float* vr = w_hh + (long)t * H;
    const float* vz = w_hh + (long)(H + t) * H;
    const float* vn = w_hh + (long)(2 * H + t) * H;
    for (int k = 0; k < H; ++k) {
        float hv = h[k];
        ghr += vr[k] * hv; ghz += vz[k] * hv; ghn += vn[k] * hv;
    }
    float r  = 1.f / (1.f + expf(-(gxr + ghr)));
    float z  = 1.f / (1.f + expf(-(gxz + ghz)));
    float nn = tanhf(gxn + r * ghn);
    state_out[b * H + t] = (1.f - z) * nn + z * h[t];
}

// ---------------- K2: score_g = state @ Wo_w^T + Wo_b (WMMA f32) ------------
// One wave per 16-col N tile; computes BOTH 16-row M tiles reusing B fragment.
__global__ void __launch_bounds__(256)
score_g_kernel(const float* __restrict__ state, const float* __restrict__ Wo_w,
               const float* __restrict__ Wo_b, float* __restrict__ score_g) {
    __shared__ float sA[B * H];                // 32x256 state, 32 KB
    stage_to_lds(state, sA, B * H);
    __syncthreads();

    const int wave = threadIdx.x >> 5;
    const int l  = threadIdx.x & 31;
    const int lh = l >> 4, ll = l & 15;
    const long ntile = (long)blockIdx.x * 8 + wave;
    if (ntile >= V / 16) return;               // wave-uniform
    const long n0 = ntile * 16;

    const float* bp = Wo_w + (n0 + ll) * (long)H + 2 * lh;
    v8f acc0 = {}; v8f acc1 = {};
    for (int k = 0; k < H; k += 4) {
        v2f bf = { bp[k], bp[k + 1] };
        v2f a0 = { sA[ll * H + k + 2 * lh],        sA[ll * H + k + 2 * lh + 1] };
        v2f a1 = { sA[(16 + ll) * H + k + 2 * lh], sA[(16 + ll) * H + k + 2 * lh + 1] };
        acc0 = wmma4(a0, bf, acc0);
        acc1 = wmma4(a1, bf, acc1);
    }
    const float bias = Wo_b[n0 + ll];
#pragma unroll
    for (int i = 0; i < 8; ++i) {
        int m = i + 8 * lh;                    // C layout: VGPR i, lane l
        score_g[(long)m * V + n0 + ll]        = acc0[i] + bias;
        score_g[(long)(16 + m) * V + n0 + ll] = acc1[i] + bias;
    }
}

// ---- K3: fused sc = tanh(encoded @ Wc_w^T + Wc_b); score_c = tanh(sc.state) -
// Block = 64 encoded rows (one b, 64 consecutive s) staged in dynamic LDS via
// async global->LDS (ASYNCcnt). 8 waves x (2 N-tiles x 4 M-tiles) accumulators;
// B fragment reused 4x across the M super-tile.
__global__ void __launch_bounds__(256)
scorec_kernel(const float* __restrict__ encoded, const int* __restrict__ encoded_idx,
              const float* __restrict__ state, const float* __restrict__ Wc_w,
              const float* __restrict__ Wc_b, float* __restrict__ score_c) {
    extern __shared__ float smem[];
    float* encA   = smem;                      // 64*512 floats = 128 KB
    float* stateS = smem + 64 * TWOH;          // 256
    float* rowsum = stateS + H;                // 64

    const int b     = blockIdx.x >> 5;         // S/64 = 32 tiles per batch row
    const int stile = blockIdx.x & 31;
    const long s0   = (long)stile * 64;

    const float* src = encoded + ((long)b * S + s0) * TWOH;   // 64 contiguous rows
    stage_to_lds(src, encA, 64 * TWOH);
    if (threadIdx.x < H)  stateS[threadIdx.x] = state[b * H + threadIdx.x];
    if (threadIdx.x < 64) rowsum[threadIdx.x] = 0.f;
    __syncthreads();

    const int wave = threadIdx.x >> 5;
    const int l  = threadIdx.x & 31;
    const int lh = l >> 4, ll = l & 15;
    const int n0 = wave * 32;                  // 2 N tiles: n0, n0+16

    const float* b0p = Wc_w + (long)(n0 + ll) * TWOH + 2 * lh;
    const float* b1p = Wc_w + (long)(n0 + 16 + ll) * TWOH + 2 * lh;

    v8f acc[8] = {};                           // [mt][j] -> acc[2*mt+j]
    for (int k = 0; k < TWOH; k += 4) {
        v2f bf0 = { b0p[k], b0p[k + 1] };
        v2f bf1 = { b1p[k], b1p[k + 1] };
#pragma unroll
        for (int mt = 0; mt < 4; ++mt) {
            const int row = mt * 16 + ll;
            v2f a = { encA[row * TWOH + k + 2 * lh], encA[row * TWOH + k + 2 * lh + 1] };
            acc[2 * mt + 0] = wmma4(a, bf0, acc[2 * mt + 0]);
            acc[2 * mt + 1] = wmma4(a, bf1, acc[2 * mt + 1]);
        }
    }
    // epilogue: tanh(+bias), dot with state, reduce per row into LDS
    const float st0 = stateS[n0 + ll],     st1 = stateS[n0 + 16 + ll];
    const float cb0 = Wc_b[n0 + ll],       cb1 = Wc_b[n0 + 16 + ll];
#pragma unroll
    for (int mt = 0; mt < 4; ++mt) {
#pragma unroll
        for (int i = 0; i < 8; ++i) {
            const int m = mt * 16 + i + 8 * lh;
            float v0 = tanhf(acc[2 * mt + 0][i] + cb0) * st0;
            float v1 = tanhf(acc[2 * mt + 1][i] + cb1) * st1;
            atomicAdd(&rowsum[m], v0 + v1);    // ds_add_f32
        }
    }
    __syncthreads();
    if (threadIdx.x < 64) {
        const long s = s0 + threadIdx.x;
        const int idx = encoded_idx[(long)b * S + s];
        score_c[(long)b * S + s] = tanhf(rowsum[threadIdx.x]) + (idx == 0 ? -1000.f : 0.f);
    }
}

// ---------------- K4: per-row softmax max & sumexp over [score_g|score_c] ---
__global__ void __launch_bounds__(256)
softmax_stats_kernel(const float* __restrict__ score_g, const float* __restrict__ score_c,
                     float* __restrict__ stats) {
    __shared__ float red[256];
    const int b = blockIdx.x;
    const float* g = score_g + (long)b * V;
    const float* c = score_c + (long)b * S;
    float m = -3.4e38f;
    for (int i = threadIdx.x; i < V; i += 256) m = fmaxf(m, g[i]);
    for (int i = threadIdx.x; i < S; i += 256) m = fmaxf(m, c[i]);
    red[threadIdx.x] = m; __syncthreads();
    for (int off = 128; off > 0; off >>= 1) {
        if (threadIdx.x < off) red[threadIdx.x] = fmaxf(red[threadIdx.x], red[threadIdx.x + off]);
        __syncthreads();
    }
    const float maxv = red[0]; __syncthreads();
    float s = 0.f;
    for (int i = threadIdx.x; i < V; i += 256) s += expf(g[i] - maxv);
    for (int i = threadIdx.x; i < S; i += 256) s += expf(c[i] - maxv);
    red[threadIdx.x] = s; __syncthreads();
    for (int off = 128; off > 0; off >>= 1) {
        if (threadIdx.x < off) red[threadIdx.x] += red[threadIdx.x + off];
        __syncthreads();
    }
    if (threadIdx.x == 0) { stats[2 * b] = maxv; stats[2 * b + 1] = red[0]; }
}

// ---------------- K5a: out = prob_g_ext (probs for v<V, 1e-4 filler) --------
__global__ void probg_kernel(const float* __restrict__ score_g,
                             const float* __restrict__ stats, float* __restrict__ out) {
    long i = (long)blockIdx.x * 256 + threadIdx.x;
    if (i >= (long)B * VEXT) return;
    int b = (int)(i / VEXT), v = (int)(i % VEXT);
    out[i] = (v < V) ? expf(score_g[(long)b * V + v] - stats[2 * b]) / stats[2 * b + 1]
                     : 1e-4f;
}

// ---------------- K5b: score_c -> prob_c (in place) -------------------------
__global__ void probc_kernel(float* __restrict__ score_c, const float* __restrict__ stats) {
    int i = blockIdx.x * 256 + threadIdx.x;    // B*S = 65536
    int b = i >> 11;
    score_c[i] = expf(score_c[i] - stats[2 * b]) / stats[2 * b + 1];
}

// ---------------- K6: global histogram of encoded_idx -----------------------
__global__ void hist_kernel(const int* __restrict__ encoded_idx, int* __restrict__ counts) {
    int i = blockIdx.x * 256 + threadIdx.x;
    atomicAdd(&counts[encoded_idx[i]], 1);
}

// ---------------- K7a: row_sums scatter-add of prob_c -----------------------
__global__ void rowsum_kernel(const int* __restrict__ encoded_idx,
                              const float* __restrict__ prob_c, float* __restrict__ row_sums) {
    int i = blockIdx.x * 256 + threadIdx.x;
    int b = i >> 11;
    atomicAdd(&row_sums[(long)b * VEXT + encoded_idx[i]], prob_c[i]);
}

// ---------------- K7b: attn scatter into out --------------------------------
__global__ void scatter_kernel(const int* __restrict__ encoded_idx,
                               const float* __restrict__ prob_c,
                               const int* __restrict__ counts,
                               const float* __restrict__ row_sums, float* __restrict__ out) {
    int i = blockIdx.x * 256 + threadIdx.x;
    int b = i >> 11;
    int idx = encoded_idx[i];
    float pc = prob_c[i];
    float attn = (counts[idx] > 1) ? pc * row_sums[(long)b * VEXT + idx] : pc;
    atomicAdd(&out[(long)b * VEXT + idx], attn);
}

// ------ K8: sparse weighted_out = sum_s attn2[b,s] * encoded[b,s,:] ---------
__global__ void __launch_bounds__(256)
weighted_kernel(const int* __restrict__ encoded_idx, const int* __restrict__ input_idx,
                const float* __restrict__ prob_c, const float* __restrict__ encoded,
                float* __restrict__ out_weighted) {
    __shared__ int list[S];
    __shared__ int cnt;
    const int b = blockIdx.x;
    if (threadIdx.x == 0) cnt = 0;
    __syncthreads();
    const int target = input_idx[b];
    for (int s = threadIdx.x; s < S; s += 256)
        if (encoded_idx[b * S + s] == target) list[atomicAdd(&cnt, 1)] = s;
    __syncthreads();
    const int n = cnt;
    const float inv = (n > 1) ? 1.0f / (float)n : 1.0f;
    float acc0 = 0.f, acc1 = 0.f;
    for (int k = 0; k < n; ++k) {
        const int s = list[k];
        const float w = prob_c[b * S + s] * inv;
        const float* e = encoded + ((long)b * S + s) * TWOH;
        acc0 += w * e[threadIdx.x];
        acc1 += w * e[threadIdx.x + H];
    }
    out_weighted[b * TWOH + threadIdx.x]     = acc0;
    out_weighted[b * TWOH + threadIdx.x + H] = acc1;
}

// ---------------- launch ----------------
extern "C" void kernel_launch(void* const* d_in, const int* in_sizes, int n_in,
                              void* d_out, int out_size, void* d_ws, size_t ws_size,
                              hipStream_t stream) {
    const int*   input_idx   = (const int*)  d_in[0];
    const float* encoded     = (const float*)d_in[1];
    const int*   encoded_idx = (const int*)  d_in[2];
    const float* prev_state  = (const float*)d_in[3];
    const float* weighted    = (const float*)d_in[4];
    // d_in[5] = order (unused)
    const float* embed       = (const float*)d_in[6];
    const float* w_ih        = (const float*)d_in[7];
    const float* w_hh        = (const float*)d_in[8];
    const float* b_ih        = (const float*)d_in[9];
    const float* b_hh        = (const float*)d_in[10];
    const float* Wo_w        = (const float*)d_in[11];
    const float* Wo_b        = (const float*)d_in[12];
    const float* Wc_w        = (const float*)d_in[13];
    const float* Wc_b        = (const float*)d_in[14];

    float* out          = (float*)d_out;                 // (B, VEXT)
    float* out_state    = out + (size_t)B * VEXT;        // (B, H)
    float* out_weighted = out_state + (size_t)B * H;     // (B, 2H)

    // workspace carve (floats): score_g | score_c | stats | counts | row_sums
    float* ws       = (float*)d_ws;
    float* score_g  = ws;                                // B*V
    float* score_c  = score_g + (size_t)B * V;           // B*S
    float* stats    = score_c + (size_t)B * S;           // 2*B
    int*   counts   = (int*)(stats + 2 * B);             // VEXT
    float* row_sums = (float*)(counts + VEXT);           // B*VEXT

    // K0: zero counts + row_sums (contiguous)
    {
        long n = (long)VEXT + (long)B * VEXT;
        zero_kernel<<<(int)((n + 255) / 256), 256, 0, stream>>>((unsigned*)counts, n);
    }
    // K1: GRU state
    gru_kernel<<<B, 256, 0, stream>>>(input_idx, embed, weighted, prev_state,
                                      w_ih, w_hh, b_ih, b_hh, out_state);
    // K2: score_g (WMMA f32)
    {
        int ntiles = V / 16;                             // 3125
        score_g_kernel<<<(ntiles + 7) / 8, 256, 0, stream>>>(out_state, Wo_w, Wo_b, score_g);
    }
    // K3: fused sc-GEMM + score_c (WMMA f32, async LDS staging)
    {
        size_t smem = (size_t)(64 * TWOH + H + 64) * sizeof(float);  // ~129.3 KB
        scorec_kernel<<<B * (S / 64), 256, smem, stream>>>(encoded, encoded_idx,
                                                           out_state, Wc_w, Wc_b, score_c);
    }
    // K4: softmax stats
    softmax_stats_kernel<<<B, 256, 0, stream>>>(score_g, score_c, stats);
    // K5a/b: probabilities
    probg_kernel<<<(int)(((long)B * VEXT + 255) / 256), 256, 0, stream>>>(score_g, stats, out);
    probc_kernel<<<B * S / 256, 256, 0, stream>>>(score_c, stats);
    // K6: global counts
    hist_kernel<<<B * S / 256, 256, 0, stream>>>(encoded_idx, counts);
    // K7: row sums then attn scatter into out
    rowsum_kernel<<<B * S / 256, 256, 0, stream>>>(encoded_idx, score_c, row_sums);
    scatter_kernel<<<B * S / 256, 256, 0, stream>>>(encoded_idx, score_c, counts, row_sums, out);
    // K8: sparse weighted_out
    weighted_kernel<<<B, 256, 0, stream>>>(encoded_idx, input_idx, score_c, encoded, out_weighted);
}